// DiceLoss_5471788335563
// MI455X (gfx1250) — compile-verified
//
#include <hip/hip_runtime.h>

typedef __attribute__((ext_vector_type(2))) float v2f;
typedef __attribute__((ext_vector_type(8))) float v8f;

#define Bn 2
#define Cn 32
#define Dn 64
#define Nn 16384                 // H*W = 128*128
#define SLICES (Bn * Cn * Dn)    // 4096
#define BCn (Bn * Cn)            // 64
#define THREADS 256

// sigmoid via hardware exp2 + rcp: sigmoid(x) = 1 / (1 + 2^(-x*log2(e)))
__device__ __forceinline__ float sigmoid_fast(float x) {
    float e = __builtin_amdgcn_exp2f(-1.4426950408889634f * x);
    return __builtin_amdgcn_rcpf(1.0f + e);
}

// Full-wave (32 lane) f32 sum using V_WMMA_F32_16X16X4_F32 with B = ones.
// A = {x, 0}: A[m,0]=x[m], A[m,2]=x[m+16]  ->  C[m,n] = x[m] + x[m+16].
// C VGPR j holds s[j] (lanes 0-15) / s[j+8] (lanes 16-31); summing the 8 C
// VGPRs then folding across the lane-16 boundary gives the total in all lanes.
// Requires EXEC all-ones (callers invoke it from all 256 threads, no divergence).
__device__ __forceinline__ float wave_sum32(float x) {
    v2f a = {x, 0.0f};
    v2f b = {1.0f, 1.0f};
    v8f c = {};
    c = __builtin_amdgcn_wmma_f32_16x16x4_f32(
            /*neg_a=*/false, a, /*neg_b=*/false, b,
            /*c_mod=*/(short)0, c, /*reuse_a=*/false, /*reuse_b=*/false);
    float r = ((c[0] + c[1]) + (c[2] + c[3])) + ((c[4] + c[5]) + (c[6] + c[7]));
    r += __shfl_xor(r, 16, 32);
    return r;
}

__global__ void __launch_bounds__(THREADS)
dice_slice_kernel(const float* __restrict__ predict,
                  const float* __restrict__ target,
                  float* __restrict__ ws_dice,
                  float* __restrict__ ws_valid) {
    const int slice = blockIdx.x;          // (b*C + c)*D + d
    const int tid   = threadIdx.x;

    // 4096 float4 per slice; 256 threads -> 16 float4 pairs per thread
    const float4* p4 = reinterpret_cast<const float4*>(predict) + ((size_t)slice << 12);
    const float4* t4 = reinterpret_cast<const float4*>(target)  + ((size_t)slice << 12);

    float num = 0.0f, sp = 0.0f, st = 0.0f;
#pragma unroll 4
    for (int it = 0; it < (Nn / 4) / THREADS; ++it) {
        const int i = it * THREADS + tid;
        const float4 p = p4[i];
        const float4 t = t4[i];
        const float s0 = sigmoid_fast(p.x);
        const float s1 = sigmoid_fast(p.y);
        const float s2 = sigmoid_fast(p.z);
        const float s3 = sigmoid_fast(p.w);
        num = fmaf(s0, t.x, num);
        num = fmaf(s1, t.y, num);
        num = fmaf(s2, t.z, num);
        num = fmaf(s3, t.w, num);
        sp += (s0 + s1) + (s2 + s3);
        st += (t.x + t.y) + (t.z + t.w);
    }

    // wave-level reduction via WMMA (all lanes active here)
    num = wave_sum32(num);
    sp  = wave_sum32(sp);
    st  = wave_sum32(st);

    __shared__ float sN[THREADS / 32], sP[THREADS / 32], sT[THREADS / 32];
    const int wave = tid >> 5;
    if ((tid & 31) == 0) { sN[wave] = num; sP[wave] = sp; sT[wave] = st; }
    __syncthreads();

    if (tid == 0) {
        float n = 0.0f, a = 0.0f, b = 0.0f;
        for (int w = 0; w < THREADS / 32; ++w) { n += sN[w]; a += sP[w]; b += sT[w]; }
        const float t0    = target[(size_t)slice << 14];           // element 0 of slice
        const float valid = (t0 != -1.0f) ? 1.0f : 0.0f;
        const float loss  = 1.0f - 2.0f * n / (a + b + 1.0f);      // SMOOTH = 1
        ws_dice[slice]  = loss * valid;
        ws_valid[slice] = valid;
    }
}

__global__ void dice_finalize_kernel(const float* __restrict__ ws_dice,
                                     const float* __restrict__ ws_valid,
                                     const int* __restrict__ organ_mask,
                                     float* __restrict__ out) {
    const int tid = threadIdx.x;           // one thread per (b,c), 64 total
    float lm = 0.0f, mm = 0.0f;
    if (tid < BCn) {
        float dl = 0.0f, vl = 0.0f;
        for (int d = 0; d < Dn; ++d) {
            dl += ws_dice[tid * Dn + d];
            vl += ws_valid[tid * Dn + d];
        }
        const float loss_bc = dl / vl;
        const float m = (float)organ_mask[tid];
        lm = loss_bc * m;
        mm = m;
    }
    __shared__ float sl[BCn], sm[BCn];
    sl[tid] = lm;
    sm[tid] = mm;
    __syncthreads();
    if (tid == 0) {
        float a = 0.0f, b = 0.0f;
        for (int i = 0; i < BCn; ++i) { a += sl[i]; b += sm[i]; }
        out[0] = a / b;
    }
}

extern "C" void kernel_launch(void* const* d_in, const int* in_sizes, int n_in,
                              void* d_out, int out_size, void* d_ws, size_t ws_size,
                              hipStream_t stream) {
    (void)in_sizes; (void)n_in; (void)out_size; (void)ws_size;
    const float* predict    = (const float*)d_in[0];
    const float* target     = (const float*)d_in[1];
    const int*   organ_mask = (const int*)d_in[2];
    float* out = (float*)d_out;

    float* ws_dice  = (float*)d_ws;          // [SLICES]
    float* ws_valid = ws_dice + SLICES;      // [SLICES]  (32 KB total, fully rewritten)

    dice_slice_kernel<<<SLICES, THREADS, 0, stream>>>(predict, target, ws_dice, ws_valid);
    dice_finalize_kernel<<<1, BCn, 0, stream>>>(ws_dice, ws_valid, organ_mask, out);
}